// hybrid_83966610637033
// MI455X (gfx1250) — compile-verified
//
#include <hip/hip_runtime.h>
#include <math.h>

// Problem constants (from the reference)
#define S_LEN  256
#define B_SZ   8
#define V_SZ   32000
#define N_ROWS (S_LEN * B_SZ)   // 2048
#define KWIN   30
#define ALPHA  0.5f

typedef __attribute__((ext_vector_type(2))) float v2f;
typedef __attribute__((ext_vector_type(4))) float v4f;
typedef __attribute__((ext_vector_type(8))) float v8f;

// ---------------------------------------------------------------------------
// Merged streaming kernel: 4096 blocks, 256 threads (8 wave32 waves).
//   blocks [0, 2048)    -> scores rows: masked NLL per row
//   blocks [2048, 4096) -> pred rows: argmax index + max softmax prob
// Branchless online softmax stats at float4 granularity:
//   5 exps / 4 elements, single pass over HBM (1x traffic), no lane divergence.
// ---------------------------------------------------------------------------
__global__ void __launch_bounds__(256)
row_stats_kernel(const float* __restrict__ scores,
                 const int*   __restrict__ gtruth,
                 const float* __restrict__ pred,
                 float*       __restrict__ nll_out,
                 float*       __restrict__ maxval_out,
                 int*         __restrict__ idx_out) {
  const bool is_pred = (blockIdx.x >= N_ROWS);          // wave-uniform
  const int  r       = is_pred ? (blockIdx.x - N_ROWS) : blockIdx.x;
  const int  tid     = threadIdx.x;
  const float* row   = (is_pred ? pred : scores) + (size_t)r * V_SZ;
  const v4f*   row4  = (const v4f*)row;

  float m = -INFINITY;
  float s = 0.0f;
  int   best = 0;

  for (int i = tid; i < V_SZ / 4; i += 256) {
    if (i + 256 < V_SZ / 4) __builtin_prefetch(row4 + i + 256, 0, 0);
    v4f v = row4[i];
    float x0 = v[0], x1 = v[1], x2 = v[2], x3 = v[3];

    // 4-wide max + first-occurrence offset (pure cndmask selects)
    float ma = x0; int oa = 0;
    if (x1 > ma) { ma = x1; oa = 1; }
    if (x2 > ma) { ma = x2; oa = 2; }
    if (x3 > ma) { ma = x3; oa = 3; }

    float mn = fmaxf(m, ma);
    s = s * __expf(m - mn)
      + __expf(x0 - mn) + __expf(x1 - mn)
      + __expf(x2 - mn) + __expf(x3 - mn);
    if (ma > m) best = i * 4 + oa;   // select; in-thread order is ascending
    m = mn;
  }

  __shared__ float sm[256], ss[256];
  __shared__ int   si[256];
  sm[tid] = m; ss[tid] = s; si[tid] = best;
  __syncthreads();
  for (int off = 128; off > 0; off >>= 1) {
    if (tid < off) {
      float m1 = sm[tid], m2 = sm[tid + off];
      float s1 = ss[tid], s2 = ss[tid + off];
      int   i1 = si[tid], i2 = si[tid + off];
      float mn = fmaxf(m1, m2);
      ss[tid] = s1 * __expf(m1 - mn) + s2 * __expf(m2 - mn);
      sm[tid] = mn;
      si[tid] = (m2 > m1 || (m2 == m1 && i2 < i1)) ? i2 : i1;  // first occurrence
    }
    __syncthreads();
  }

  if (tid == 0) {
    if (is_pred) {
      maxval_out[r] = 1.0f / ss[0];            // max softmax prob = exp(m-m)/sumexp
      idx_out[r]    = si[0];
    } else {
      int   tgt = gtruth[r];
      float tl  = row[tgt];
      float nll = (sm[0] + __logf(ss[0])) - tl; // -log_softmax[target]
      nll_out[r] = (tgt != 0) ? nll : 0.0f;
    }
  }
}

// ---------------------------------------------------------------------------
// WMMA-based sum of 256 floats in LDS (valid in all lanes on return).
// V_WMMA_F32_16X16X4_F32 with all-ones B; C accumulates four 16x4 A-chunks.
// Every C element holds a row-sum replicated 16x across N -> total = sum(C)/16.
// Must be executed with EXEC all-ones: called uniformly by all 256 threads.
// ---------------------------------------------------------------------------
__device__ __forceinline__ float wmma_sum256(const float* x) {
  const int lane = threadIdx.x & 31;
  const int mrow = lane & 15;
  const int kb   = (lane < 16) ? 0 : 2;  // ISA A-layout: lanes 0-15 -> K=0,1; 16-31 -> K=2,3
  v8f c = {};
  v2f b; b.x = 1.0f; b.y = 1.0f;
#pragma unroll
  for (int ch = 0; ch < 4; ++ch) {
    v2f a;
    a.x = x[ch * 64 + mrow * 4 + kb];
    a.y = x[ch * 64 + mrow * 4 + kb + 1];
    c = __builtin_amdgcn_wmma_f32_16x16x4_f32(false, a, false, b,
                                              (short)0, c, false, false);
  }
  float t = c[0] + c[1] + c[2] + c[3] + c[4] + c[5] + c[6] + c[7];
#pragma unroll
  for (int off = 1; off < 32; off <<= 1) t += __shfl_xor(t, off, 32);
  return t * (1.0f / 16.0f);
}

// ---------------------------------------------------------------------------
// Finalize: banded duplicate count + scalar combine, single block.
// ---------------------------------------------------------------------------
__global__ void __launch_bounds__(256)
finalize_kernel(const float* __restrict__ nll,
                const int*   __restrict__ gtruth,
                const float* __restrict__ maxval,
                const int*   __restrict__ idx,
                float*       __restrict__ out) {
  __shared__ float pn[256], pc[256], pl[256];
  const int tid = threadIdx.x;

  float an = 0.0f, ac = 0.0f, al = 0.0f;
  for (int p = tid; p < N_ROWS; p += 256) {
    an += nll[p];
    ac += (gtruth[p] != 0) ? 1.0f : 0.0f;
    const int i  = p / B_SZ;
    const int b  = p % B_SZ;
    const int my = idx[p];
    const int j0 = (i >= KWIN) ? (i - KWIN) : 0;
    int cnt = 0;
    for (int j = j0; j < i; ++j) cnt += (idx[j * B_SZ + b] == my) ? 1 : 0;
    al += maxval[p] * (float)cnt;
  }
  pn[tid] = an; pc[tid] = ac; pl[tid] = al;
  __syncthreads();

  // All waves run the WMMA reduction uniformly (EXEC all-ones requirement);
  // every wave computes the same totals, thread 0 writes the result.
  float tn = wmma_sum256(pn);
  float tc = wmma_sum256(pc);
  float tl = wmma_sum256(pl);

  if (tid == 0) {
    float cross = tn / tc;                 // mean NLL over targets != 0
    float res   = (1.0f - ALPHA) * cross;
    if (tl != 0.0f) res += ALPHA * tl;     // add local term only if nonzero
    out[0] = res;
  }
}

// ---------------------------------------------------------------------------
extern "C" void kernel_launch(void* const* d_in, const int* in_sizes, int n_in,
                              void* d_out, int out_size, void* d_ws, size_t ws_size,
                              hipStream_t stream) {
  const float* scores = (const float*)d_in[0];
  const int*   gtruth = (const int*)d_in[1];
  const float* pred   = (const float*)d_in[2];

  float* ws        = (float*)d_ws;
  float* ws_nll    = ws;                      // [2048] f32
  float* ws_maxval = ws + N_ROWS;             // [2048] f32
  int*   ws_idx    = (int*)(ws + 2 * N_ROWS); // [2048] i32   (24 KB total)

  row_stats_kernel<<<2 * N_ROWS, 256, 0, stream>>>(scores, gtruth, pred,
                                                   ws_nll, ws_maxval, ws_idx);
  finalize_kernel<<<1, 256, 0, stream>>>(ws_nll, gtruth, ws_maxval, ws_idx,
                                         (float*)d_out);
}